// RNN_10591389352188
// MI455X (gfx1250) — compile-verified
//
#include <hip/hip_runtime.h>
#include <math.h>

// Problem constants
#define BB    64
#define TT    2048
#define II    14
#define HH    100
#define LL    10
#define NP    112   // padded H (7 * 16)
#define KP    100   // K depth for hidden GEMM (25 k-tiles of 4)
#define NKT   25    // K tiles per GEMM
#define K0T   4     // K tiles for layer-0 input GEMM (K=16)
#define S     116   // LDS row stride for activations (conflict-free, float4 aligned)
#define DEPTH 4     // inter-layer ring depth (power of 2)

// Swizzled weight block: [NKT][7][32 lanes][2] floats = 11200 floats per matrix.
#define WSWZ  (NKT * 7 * 32 * 2)

// Workspace layout (floats):
//   [0 .. 63]   flags (prod[l] at int index l, cons[l] at 16+l)
//   [64 ..)     hbuf   : 9 * DEPTH * BB * NP inter-layer ring buffers
//   then        wpack  : per layer {wih_swz, whh_swz} in WMMA B-fragment order
//   then        biaspk : LL * NP  (b_ih + b_hh, padded)
#define HBUF_OFF  64
#define HBUF_SZ   (9 * DEPTH * BB * NP)
#define WL_SZ     (2 * WSWZ)
#define WPACK_OFF (HBUF_OFF + HBUF_SZ)
#define WPACK_SZ  (LL * WL_SZ)
#define BIAS_OFF  (WPACK_OFF + WPACK_SZ)

// LDS layout (floats): 37,472 f = 149,888 B  (<< 320 KB WGP LDS)
#define L_XIN   0
#define L_HSELF (L_XIN + BB * S)            //  7424
#define L_WIH   (L_HSELF + BB * S)          // 14848
#define L_WHH   (L_WIH + WSWZ)              // 26048
#define L_BIAS  (L_WHH + WSWZ)              // 37248
#define L_FCW   (L_BIAS + NP)               // 37360
#define L_TOTAL (L_FCW + NP)                // 37472

typedef __attribute__((ext_vector_type(2))) float v2f;
typedef __attribute__((ext_vector_type(8))) float v8f;

// Branch-free tanh on hardware transcendentals: ~1 ulp-class, saturates for
// large |x| (exp2 underflows -> r = 1), preserves sign/zero.
__device__ __forceinline__ float fast_tanh(float x) {
  const float ax = __builtin_fabsf(x);
  const float e  = __builtin_amdgcn_exp2f(ax * -2.885390081777927f); // e^(-2|x|)
  const float r  = (1.0f - e) * __builtin_amdgcn_rcpf(1.0f + e);
  return __builtin_copysignf(r, x);
}

// ---------------------------------------------------------------------------
// Init kernel: zero sync flags; pack weights directly in WMMA B-fragment
// order: element (ktile, n, lane, j) = W^T[4*ktile + 2*(lane>=16) + j][16n + lane%16].
// One contiguous ds_load_b64 per fragment in the pipeline's inner loop.
// ---------------------------------------------------------------------------
__global__ void rnn_init(const float* __restrict__ w_ih0,
                         const float* __restrict__ w_ihL,
                         const float* __restrict__ w_hh,
                         const float* __restrict__ b_ih,
                         const float* __restrict__ b_hh,
                         float* __restrict__ ws) {
  const int tid = blockIdx.x * blockDim.x + threadIdx.x;
  const int nth = gridDim.x * blockDim.x;
  if (blockIdx.x == 0 && threadIdx.x < 64) ((int*)ws)[threadIdx.x] = 0;

  for (int idx = tid; idx < WPACK_SZ; idx += nth) {
    const int l     = idx / WL_SZ;
    const int r     = idx % WL_SZ;
    const int which = r / WSWZ;             // 0 = input weights, 1 = hidden weights
    const int q     = r % WSWZ;
    const int kt    = q / (7 * 64);
    const int rem   = q % (7 * 64);
    const int n     = rem / 64;
    const int lane  = (rem % 64) >> 1;
    const int j     = rem & 1;
    const int k     = kt * 4 + ((lane >> 4) << 1) + j;
    const int col   = n * 16 + (lane & 15);
    float v = 0.0f;
    if (col < HH) {
      if (which == 0) {
        if (l == 0) { if (k < II) v = w_ih0[col * II + k]; }
        else        { if (k < HH) v = w_ihL[((l - 1) * HH + col) * HH + k]; }
      } else {
        if (k < HH) v = w_hh[(l * HH + col) * HH + k];
      }
    }
    ws[WPACK_OFF + idx] = v;
  }
  for (int idx = tid; idx < LL * NP; idx += nth) {
    const int l = idx / NP, n = idx % NP;
    ws[BIAS_OFF + idx] = (n < HH) ? (b_ih[l * HH + n] + b_hh[l * HH + n]) : 0.0f;
  }
}

// ---------------------------------------------------------------------------
// Pipeline kernel: one workgroup (128 thr / 4 waves) per RNN layer.
// Weights LDS-resident in fragment order; hidden GEMM first to hide the
// inter-layer handoff; acquire/release ring buffers between layers.
// ---------------------------------------------------------------------------
__global__ void __launch_bounds__(128) rnn_pipeline(
    const float* __restrict__ x,      // [B, T, I]
    const float* __restrict__ hidden, // [L, B, H]
    const float* __restrict__ fc_w,   // [H]  (O = 1)
    const float* __restrict__ fc_b,   // [1]
    float* __restrict__ out,          // [B*T] then [L,B,H]
    float* __restrict__ ws) {
  extern __shared__ float smem[];
  float* xin   = smem + L_XIN;
  float* hself = smem + L_HSELF;
  float* wihs  = smem + L_WIH;
  float* whhs  = smem + L_WHH;
  float* biasl = smem + L_BIAS;
  float* fcw   = smem + L_FCW;

  const int l    = blockIdx.x;
  const int tid  = threadIdx.x;
  const int lane = tid & 31;
  const int wave = tid >> 5;

  int*   prod = (int*)ws;
  int*   cons = (int*)ws + 16;
  float* hbuf = ws + HBUF_OFF;
  const float* wpk = ws + WPACK_OFF + l * WL_SZ;    // both matrices, contiguous
  const float* bsg = ws + BIAS_OFF + l * NP;

  // ---- one-time LDS population ----
  for (int i = tid; i < BB * S; i += 128) xin[i] = 0.0f;
  for (int i = tid; i < BB * S; i += 128) {
    const int b = i / S, c = i % S;
    hself[i] = (c < HH) ? hidden[(l * BB + b) * HH + c] : 0.0f;
  }
  for (int idx = tid; idx < WL_SZ / 4; idx += 128)          // verbatim copy
    ((float4*)wihs)[idx] = ((const float4*)wpk)[idx];
  for (int i = tid; i < NP; i += 128) {
    biasl[i] = bsg[i];
    fcw[i]   = (i < HH) ? fc_w[i] : 0.0f;
  }
  const float fcb = fc_b[0];
  __syncthreads();

  // Per-lane fragment geometry (wave32 WMMA 16x16 layouts).
  const int m0   = wave * 16;          // wave's M-tile
  const int hl   = lane & 15;
  const int kq   = (lane >> 4) * 2;    // lanes 16-31 hold K+2, K+3
  const int rowA = m0 + hl;
  const int rhi  = (lane >> 4) * 8;    // C/D: lanes 16-31 hold M = r + 8

  for (int t = 0; t < TT; ++t) {
    // ---- accumulators seeded with (b_ih + b_hh) ----
    v8f acc[7];
#pragma unroll
    for (int n = 0; n < 7; ++n) {
      const float bv = biasl[n * 16 + hl];
      v8f av;
#pragma unroll
      for (int r = 0; r < 8; ++r) av[r] = bv;
      acc[n] = av;
    }

    // ---- hidden GEMM first (wave-private data; hides the handoff wait) ----
    for (int kt = 0; kt < NKT; ++kt) {
      const int kr = kt * 4 + kq;
      const v2f a = *(const v2f*)&hself[rowA * S + kr];
      const float* wb = whhs + (kt * 7 * 32 + lane) * 2;
      v2f bf[7];
#pragma unroll
      for (int n = 0; n < 7; ++n) bf[n] = *(const v2f*)(wb + n * 64);
#pragma unroll
      for (int n = 0; n < 7; ++n)
        acc[n] = __builtin_amdgcn_wmma_f32_16x16x4_f32(
            false, a, false, bf[n], (short)0, acc[n], false, false);
    }

    // ---- acquire this step's input into LDS ----
    if (l == 0) {
      for (int idx = tid; idx < BB * II; idx += 128) {
        const int b = idx / II, i = idx % II;
        xin[b * S + i] = x[(b * TT + t) * II + i];
      }
    } else {
      while (__hip_atomic_load(&prod[l - 1], __ATOMIC_ACQUIRE,
                               __HIP_MEMORY_SCOPE_AGENT) < t + 1)
        __builtin_amdgcn_s_sleep(2);
      const float* src = hbuf + (((l - 1) * DEPTH + (t & (DEPTH - 1))) * BB) * NP;
      for (int idx = tid; idx < BB * (NP / 4); idx += 128) {
        const int b = idx / (NP / 4), j = idx % (NP / 4);
        ((float4*)(xin + b * S))[j] = ((const float4*)(src + b * NP))[j];
      }
    }
    __syncthreads();
    if (l >= 1 && tid == 0)
      __hip_atomic_store(&cons[l], t + 1, __ATOMIC_RELEASE, __HIP_MEMORY_SCOPE_AGENT);

    // ---- input GEMM ----
    const int NKIN = (l == 0) ? K0T : NKT;
    for (int kt = 0; kt < NKIN; ++kt) {
      const int kr = kt * 4 + kq;
      const v2f a = *(const v2f*)&xin[rowA * S + kr];
      const float* wb = wihs + (kt * 7 * 32 + lane) * 2;
      v2f bf[7];
#pragma unroll
      for (int n = 0; n < 7; ++n) bf[n] = *(const v2f*)(wb + n * 64);
#pragma unroll
      for (int n = 0; n < 7; ++n)
        acc[n] = __builtin_amdgcn_wmma_f32_16x16x4_f32(
            false, a, false, bf[n], (short)0, acc[n], false, false);
    }

    // ---- back-pressure before overwriting ring slot ----
    if (l < LL - 1 && t >= DEPTH) {
      while (__hip_atomic_load(&cons[l + 1], __ATOMIC_ACQUIRE,
                               __HIP_MEMORY_SCOPE_AGENT) < t - DEPTH + 1)
        __builtin_amdgcn_s_sleep(2);
    }

    // ---- tanh, write new hidden to LDS and (l<9) the ring buffer ----
    float* dst = hbuf + ((l * DEPTH + (t & (DEPTH - 1))) * BB) * NP;
#pragma unroll
    for (int n = 0; n < 7; ++n) {
#pragma unroll
      for (int r = 0; r < 8; ++r) {
        const float v = fast_tanh(acc[n][r]);
        const int row = m0 + r + rhi;
        const int col = n * 16 + hl;
        hself[row * S + col] = v;
        if (l < LL - 1) dst[row * NP + col] = v;
      }
    }
    __threadfence();
    __syncthreads();

    if (l < LL - 1) {
      if (tid == 0)
        __hip_atomic_store(&prod[l], t + 1, __ATOMIC_RELEASE, __HIP_MEMORY_SCOPE_AGENT);
    } else {
      // Fused FC head (O = 1): out[b*T + t] = fc_w . h9[b] + fc_b
      if (tid < BB) {
        float s = fcb;
        for (int h = 0; h < HH; ++h) s += fcw[h] * hself[tid * S + h];
        out[tid * TT + t] = s;
      }
    }
  }

  // ---- final hidden state: d_out[B*T + l*B*H + ...] ----
  __syncthreads();
  for (int idx = tid; idx < BB * HH; idx += 128) {
    const int b = idx / HH, h = idx % HH;
    out[BB * TT + (l * BB + b) * HH + h] = hself[b * S + h];
  }
}

extern "C" void kernel_launch(void* const* d_in, const int* in_sizes, int n_in,
                              void* d_out, int out_size, void* d_ws, size_t ws_size,
                              hipStream_t stream) {
  (void)in_sizes; (void)n_in; (void)out_size; (void)ws_size;
  const float* x      = (const float*)d_in[0];
  const float* hidden = (const float*)d_in[1];
  const float* w_ih0  = (const float*)d_in[2];
  const float* w_ihL  = (const float*)d_in[3];
  const float* w_hh   = (const float*)d_in[4];
  const float* b_ih   = (const float*)d_in[5];
  const float* b_hh   = (const float*)d_in[6];
  const float* fc_w   = (const float*)d_in[7];
  const float* fc_b   = (const float*)d_in[8];
  float* out = (float*)d_out;
  float* ws  = (float*)d_ws;

  rnn_init<<<dim3(64), dim3(256), 0, stream>>>(w_ih0, w_ihL, w_hh, b_ih, b_hh, ws);

  const size_t smem = (size_t)L_TOTAL * sizeof(float);  // 149,888 B
  rnn_pipeline<<<dim3(LL), dim3(128), smem, stream>>>(x, hidden, fc_w, fc_b, out, ws);
}